// SciBert_FastText_BiLSTM_CRF_50603304681490
// MI455X (gfx1250) — compile-verified
//
#include <hip/hip_runtime.h>

// Problem constants (from reference setup_inputs)
#define B_    64
#define S_    512
#define DB_   768
#define DW_   100
#define W_    512
#define DOUT  (DB_ + DW_)   // 868 floats per output row
#define ROW4  (DOUT / 4)    // 217 float4 per output row (3472 B, 16B-aligned stride)
#define BERT4 (DB_ / 4)     // 192 float4 bert segment
#define W2V4  (DW_ / 4)     // 25 float4 w2v segment (400 B row stride, 16B-aligned)

typedef float f32x4 __attribute__((ext_vector_type(4)));

// ---------------------------------------------------------------------------
// HOT KERNEL (placed first so the disasm snippet shows its codegen).
// One 224-thread block (7 wave32) per output row (b, s).
//   t in [0,192):   out[b,s, 4t..4t+3]  = valid ? bert_x[b,s,4t..] : 0
//   t in [192,217): out[b,s, 768+4u..]  = valid ? w2v[b, j, 4u..]  : 0
// bert: read-once stream  -> non-temporal load  (th:TH_LOAD_NT)
// w2v : gathered/reused   -> temporal load      (stays in 192MB L2)
// out : write-once stream -> non-temporal store (th:TH_STORE_NT)
// All accesses are exact 16B-aligned b128 transactions; no tail.
// ---------------------------------------------------------------------------
__global__ __launch_bounds__(224) void fuse_gather_concat_kernel(
    const float* __restrict__ bert_x,
    const float* __restrict__ w2v,
    const int*   __restrict__ mappings,
    const int*   __restrict__ first_inv,
    float*       __restrict__ out)
{
    const int row = blockIdx.x;          // 0 .. B*S-1
    const int s   = row & (S_ - 1);      // S_ = 512 (power of two)
    const int b   = row >> 9;
    const int t   = threadIdx.x;

    // Uniform per-block predicate (scalar loads): row 0 and post-break rows -> 0
    const bool valid = (s >= 1) && ((s - 1) < first_inv[b]);

    f32x4 val = {0.f, 0.f, 0.f, 0.f};
    f32x4* __restrict__ orow = (f32x4*)(out + (size_t)row * DOUT);

    if (t < BERT4) {
        if (valid) {
            const f32x4* brow = (const f32x4*)(bert_x + (size_t)row * DB_);
            val = __builtin_nontemporal_load(brow + t);      // streamed, NT
        }
        __builtin_nontemporal_store(val, orow + t);          // write-once, NT
    } else if (t < ROW4) {
        if (valid) {
            int j = mappings[b * (S_ - 1) + (s - 1)];        // uniform -> s_load
            j = j < 0 ? 0 : j;                               // jnp.clip(..., 0)
            const f32x4* wrow = (const f32x4*)(w2v + ((size_t)b * W_ + (size_t)j) * DW_);
            val = wrow[t - BERT4];                           // reused table: temporal load
        }
        __builtin_nontemporal_store(val, orow + t);          // write-once, NT
    }
}

// ---------------------------------------------------------------------------
// Kernel 2 (tiny): per-batch first index where mappings[b,i] == -1 (511 if none).
// One wave32 per batch; ballot over 32-element chunks, first set bit wins.
// valid[b, s-1] in the reference == (s-1 < first_invalid[b]).
// ---------------------------------------------------------------------------
__global__ __launch_bounds__(32) void first_invalid_kernel(
    const int* __restrict__ mappings, int* __restrict__ first_inv)
{
    const int b    = blockIdx.x;
    const int lane = threadIdx.x;          // 0..31
    const int* m   = mappings + b * (S_ - 1);

    int result = S_ - 1;                   // 511 == "no -1 anywhere"
    // 511 elements -> 16 chunks of 32 (last chunk has 31 valid lanes)
    for (int base = 0; base < S_ - 1; base += 32) {
        const int i  = base + lane;
        const bool hit = (i < S_ - 1) && (m[i] == -1);
        const unsigned long long mask = __ballot(hit);       // wave32: low 32 bits
        if (mask != 0ull) {
            result = base + (__ffsll(mask) - 1);
            break;                                            // uniform branch
        }
    }
    if (lane == 0) first_inv[b] = result;
}

// ---------------------------------------------------------------------------
// Harness entry point
//   d_in[0] bert_x   [64,512,768] f32
//   d_in[1] w2v      [64,512,100] f32
//   d_in[2] mappings [64,511]     i32
//   d_out            [64,512,868] f32
//   d_ws: first 64 ints = first_invalid per batch (rewritten every call)
// ---------------------------------------------------------------------------
extern "C" void kernel_launch(void* const* d_in, const int* in_sizes, int n_in,
                              void* d_out, int out_size, void* d_ws, size_t ws_size,
                              hipStream_t stream) {
    (void)in_sizes; (void)n_in; (void)out_size; (void)ws_size;

    const float* bert = (const float*)d_in[0];
    const float* w2v  = (const float*)d_in[1];
    const int*   maps = (const int*)d_in[2];
    int* first_inv    = (int*)d_ws;

    first_invalid_kernel<<<B_, 32, 0, stream>>>(maps, first_inv);
    fuse_gather_concat_kernel<<<B_ * S_, 224, 0, stream>>>(
        bert, w2v, maps, first_inv, (float*)d_out);
}